// gpu_fp_model_knn_2697239462397
// MI455X (gfx1250) — compile-verified
//
#include <hip/hip_runtime.h>
#include <hip/hip_bf16.h>

// ---------------------------------------------------------------- constants
#define BB   2
#define HH   64
#define WW   1024
#define NN   (HH * WW)          // 65536
#define C1C  64
#define C2C  128
#define CIN  (C1C + C2C)        // 192
#define COUT 128
#define NSS  3
#define DIST2 10000.0f
#define BN_EPS 1e-5f
#define TOTAL_PIX (BB * NN)     // 131072

typedef __attribute__((ext_vector_type(2)))  float v2f;
typedef __attribute__((ext_vector_type(8)))  float v8f;

// ------------------------------------------------------------------- utils
__global__ void zero_f32(float* __restrict__ p, int n) {
  int i = blockIdx.x * blockDim.x + threadIdx.x;
  if (i < n) p[i] = 0.0f;
}

// ----------------------------------------------------- KNN + interp + concat
// One wave32 per pixel. All lanes redundantly run the 25-tap scan (uniform),
// then lanes stripe over channels for coalesced gathers/stores.
// Writes X0 row (CIN=192 floats): [feat1(64) | interp(128)].
__global__ __launch_bounds__(128) void knn_interp_concat(
    const float* __restrict__ xyz1, const float* __restrict__ xyz2,
    const float* __restrict__ feat1, const float* __restrict__ feat2,
    float* __restrict__ X0) {
  const int pix  = blockIdx.x * 4 + (threadIdx.x >> 5);
  const int lane = threadIdx.x & 31;
  if (pix >= TOTAL_PIX) return;
  const int b = pix / NN;
  const int i = pix - b * NN;
  const int h = i / WW;
  const int w = i - h * WW;

  const float* p1 = xyz1 + ((size_t)b * NN + i) * 3;
  const float p1x = p1[0], p1y = p1[1], p1z = p1[2];

  int   nidx[NSS];
  float d[NSS];
  int   cnt = 0;
  #pragma unroll
  for (int t = 0; t < 25; ++t) {
    const int dy = t / 5 - 2;
    const int dx = t % 5 - 2;
    const int nh = h + dy, nw = w + dx;
    if (nh < 0 || nh >= HH || nw < 0 || nw >= WW) continue;
    if (cnt >= NSS) continue;          // first-NS in scan order
    const int j = nh * WW + nw;
    const float* p2 = xyz2 + ((size_t)b * NN + j) * 3;
    const float ex = p2[0] - p1x, ey = p2[1] - p1y, ez = p2[2] - p1z;
    const float sq = ex * ex + ey * ey + ez * ez;
    if (sq < DIST2) { nidx[cnt] = j; d[cnt] = sq; ++cnt; }
  }

  // Not-found slots: grouped_xyz==0 -> dist = |xyz1|^2 (feeds the denominator
  // only; their feature contribution is masked to zero).
  const float selfd = p1x * p1x + p1y * p1y + p1z * p1z;
  float wgt[NSS];
  float wsum = 0.0f;
  #pragma unroll
  for (int s = 0; s < NSS; ++s) {
    float ds = (s < cnt) ? d[s] : selfd;
    ds = fmaxf(ds, 1e-10f);
    wgt[s] = 1.0f / ds;
    wsum  += wgt[s];
  }
  const float inv = 1.0f / wsum;
  #pragma unroll
  for (int s = 0; s < NSS; ++s) wgt[s] = (s < cnt) ? wgt[s] * inv : 0.0f;

  const size_t row = (size_t)pix * CIN;
  const float* f1 = feat1 + ((size_t)b * NN + i) * C1C;
  for (int c = lane; c < C1C; c += 32) X0[row + c] = f1[c];

  const float* f2b = feat2 + (size_t)b * NN * C2C;
  for (int c = lane; c < C2C; c += 32) {
    float acc = 0.0f;
    #pragma unroll
    for (int s = 0; s < NSS; ++s)
      if (s < cnt) acc += f2b[(size_t)nidx[s] * C2C + c] * wgt[s];
    X0[row + C1C + c] = acc;
  }
}

// ------------------------------------------------------------- WMMA GEMM
// Y[M x 128] = act(X)[M x K] @ Wm[K x 128] + bias, where act is identity
// (BN_IN=0) or BN+ReLU of the previous layer applied while staging A in LDS.
// Also accumulates per-channel sum / sumsq of Y into global stats (BN of the
// CURRENT layer), via shfl + one atomicAdd per column per block.
// Block = 256 thr = 8 waves. Block owns a 64-row strip; wave w owns N-tile
// [16w,16w+16) and 4 M sub-tiles. k-loop outermost: each B fragment feeds 4
// v_wmma_f32_16x16x4_f32 ops.
template <int K, int BN_IN>
__global__ __launch_bounds__(256) void gemm_bias_stats(
    const float* __restrict__ X, const float* __restrict__ Wm,
    const float* __restrict__ bias,
    const float* __restrict__ mu_in, const float* __restrict__ rs_in,
    const float* __restrict__ g_in, const float* __restrict__ be_in,
    float* __restrict__ Y,
    float* __restrict__ sum_out, float* __restrict__ sq_out) {
  constexpr int ROWS = 64;
  __shared__ float As[ROWS * K];         // 48KB (K=192) / 32KB (K=128)
  const int mBase = blockIdx.x * ROWS;

  // ---- stage A strip (b128 loads), optionally fusing BN+ReLU of prev layer
  const float4* Xv  = (const float4*)(X + (size_t)mBase * K);
  float4*       Asv = (float4*)As;
  for (int t = threadIdx.x; t < ROWS * K / 4; t += 256) {
    float4 v = Xv[t];
    if (BN_IN) {
      const int c = (t * 4) % K;
      v.x = fmaxf((v.x - mu_in[c    ]) * rs_in[c    ] * g_in[c    ] + be_in[c    ], 0.0f);
      v.y = fmaxf((v.y - mu_in[c + 1]) * rs_in[c + 1] * g_in[c + 1] + be_in[c + 1], 0.0f);
      v.z = fmaxf((v.z - mu_in[c + 2]) * rs_in[c + 2] * g_in[c + 2] + be_in[c + 2], 0.0f);
      v.w = fmaxf((v.w - mu_in[c + 3]) * rs_in[c + 3] * g_in[c + 3] + be_in[c + 3], 0.0f);
    }
    Asv[t] = v;
  }
  __syncthreads();

  const int wave = threadIdx.x >> 5;     // N tile id (0..7)
  const int lane = threadIdx.x & 31;
  const int n0   = wave * 16;
  const int n    = lane & 15;
  const int half = lane >> 4;            // lane group: 0 or 1
  const int m    = lane & 15;
  const int kb   = half * 2;             // K-pair per lane half (f32 A layout)

  v8f acc0 = {}, acc1 = {}, acc2 = {}, acc3 = {};
  for (int k0 = 0; k0 < K; k0 += 4) {
    v2f bv;
    bv.x = Wm[(size_t)(k0 + kb) * COUT + n0 + n];
    bv.y = Wm[(size_t)(k0 + kb + 1) * COUT + n0 + n];
    v2f a;
    a.x = As[(0 * 16 + m) * K + k0 + kb];
    a.y = As[(0 * 16 + m) * K + k0 + kb + 1];
    acc0 = __builtin_amdgcn_wmma_f32_16x16x4_f32(false, a, false, bv, (short)0, acc0, false, false);
    a.x = As[(1 * 16 + m) * K + k0 + kb];
    a.y = As[(1 * 16 + m) * K + k0 + kb + 1];
    acc1 = __builtin_amdgcn_wmma_f32_16x16x4_f32(false, a, false, bv, (short)0, acc1, false, false);
    a.x = As[(2 * 16 + m) * K + k0 + kb];
    a.y = As[(2 * 16 + m) * K + k0 + kb + 1];
    acc2 = __builtin_amdgcn_wmma_f32_16x16x4_f32(false, a, false, bv, (short)0, acc2, false, false);
    a.x = As[(3 * 16 + m) * K + k0 + kb];
    a.y = As[(3 * 16 + m) * K + k0 + kb + 1];
    acc3 = __builtin_amdgcn_wmma_f32_16x16x4_f32(false, a, false, bv, (short)0, acc3, false, false);
  }

  // ---- epilogue: bias, store, per-column stats partials
  // C/D layout: VGPR r -> row (sub*16 + r + half*8), col = n0 + n.
  const float bval = bias[n0 + n];
  float lsum = 0.0f, lsq = 0.0f;
  v8f accs[4] = {acc0, acc1, acc2, acc3};
  #pragma unroll
  for (int sub = 0; sub < 4; ++sub) {
    #pragma unroll
    for (int r = 0; r < 8; ++r) {
      const int mRow = mBase + sub * 16 + r + half * 8;
      const float v = accs[sub][r] + bval;
      Y[(size_t)mRow * COUT + n0 + n] = v;
      lsum += v;
      lsq  += v * v;
    }
  }
  // lanes L and L^16 hold the same column (different 8-row halves)
  lsum += __shfl_xor(lsum, 16, 32);
  lsq  += __shfl_xor(lsq, 16, 32);
  if (half == 0) {
    atomicAdd(&sum_out[n0 + n], lsum);
    atomicAdd(&sq_out[n0 + n],  lsq);
  }
}

__global__ __launch_bounds__(128) void bn_finalize(
    const float* __restrict__ sum, const float* __restrict__ sumsq,
    float* __restrict__ mu, float* __restrict__ rs, float invM) {
  const int c = threadIdx.x;
  const float m = sum[c] * invM;
  const float v = sumsq[c] * invM - m * m;   // population variance
  mu[c] = m;
  rs[c] = rsqrtf(v + BN_EPS);
}

__global__ __launch_bounds__(256) void bn_relu_out(
    const float* __restrict__ Y, const float* __restrict__ mu,
    const float* __restrict__ rs, const float* __restrict__ g,
    const float* __restrict__ be, float* __restrict__ out, size_t total4) {
  const size_t t = (size_t)blockIdx.x * blockDim.x + threadIdx.x;
  if (t >= total4) return;
  const int c = (int)((t * 4) & (COUT - 1));
  float4 v = ((const float4*)Y)[t];
  v.x = fmaxf((v.x - mu[c    ]) * rs[c    ] * g[c    ] + be[c    ], 0.0f);
  v.y = fmaxf((v.y - mu[c + 1]) * rs[c + 1] * g[c + 1] + be[c + 1], 0.0f);
  v.z = fmaxf((v.z - mu[c + 2]) * rs[c + 2] * g[c + 2] + be[c + 2], 0.0f);
  v.w = fmaxf((v.w - mu[c + 3]) * rs[c + 3] * g[c + 3] + be[c + 3], 0.0f);
  ((float4*)out)[t] = v;
}

// ------------------------------------------------------------------ launch
extern "C" void kernel_launch(void* const* d_in, const int* in_sizes, int n_in,
                              void* d_out, int out_size, void* d_ws, size_t ws_size,
                              hipStream_t stream) {
  const float* xyz1  = (const float*)d_in[0];
  const float* xyz2  = (const float*)d_in[1];
  const float* feat1 = (const float*)d_in[2];
  const float* feat2 = (const float*)d_in[3];
  const float* w0  = (const float*)d_in[4];
  const float* b0  = (const float*)d_in[5];
  const float* g0  = (const float*)d_in[6];
  const float* be0 = (const float*)d_in[7];
  const float* w1  = (const float*)d_in[8];
  const float* b1  = (const float*)d_in[9];
  const float* g1  = (const float*)d_in[10];
  const float* be1 = (const float*)d_in[11];

  float* ws = (float*)d_ws;
  const size_t X0_elems = (size_t)TOTAL_PIX * CIN;   // 25,165,824
  const size_t Y0_elems = (size_t)TOTAL_PIX * COUT;  // 16,777,216
  float* X0 = ws;                        // consumed by GEMM1
  float* Y0 = ws + X0_elems;             // layer-0 pre-BN output
  float* Y1 = ws;                        // reuses X0 region (fits: Y0 < X0)
  float* st = ws + X0_elems + Y0_elems;  // 1024 floats of stats
  // layer0: st[0..127]=sum  st[128..255]=sumsq  st[256..383]=mu  st[384..511]=rs
  // layer1: st[512..639]    st[640..767]        st[768..895]     st[896..1023]

  const float invM = 1.0f / (float)TOTAL_PIX;

  zero_f32<<<4, 256, 0, stream>>>(st, 1024);

  knn_interp_concat<<<TOTAL_PIX / 4, 128, 0, stream>>>(xyz1, xyz2, feat1, feat2, X0);

  // ---- layer 0: 192 -> 128 (stats fused into GEMM epilogue)
  gemm_bias_stats<CIN, 0><<<TOTAL_PIX / 64, 256, 0, stream>>>(
      X0, w0, b0, nullptr, nullptr, nullptr, nullptr, Y0, st, st + 128);
  bn_finalize<<<1, 128, 0, stream>>>(st, st + 128, st + 256, st + 384, invM);

  // ---- layer 1: 128 -> 128 (BN0+ReLU fused into A staging; stats fused)
  gemm_bias_stats<COUT, 1><<<TOTAL_PIX / 64, 256, 0, stream>>>(
      Y0, w1, b1, st + 256, st + 384, g0, be0, Y1, st + 512, st + 640);
  bn_finalize<<<1, 128, 0, stream>>>(st + 512, st + 640, st + 768, st + 896, invM);

  // ---- final BN1 + ReLU -> output
  bn_relu_out<<<(int)(Y0_elems / 4 / 256), 256, 0, stream>>>(
      Y1, st + 768, st + 896, g1, be1, (float*)d_out, Y0_elems / 4);
}